// ApplyTimeChannel_69587060129824
// MI455X (gfx1250) — compile-verified
//
#include <hip/hip_runtime.h>
#include <stdint.h>

// ---------------- problem constants (from reference) ----------------
#define B_    32
#define RXA_  16
#define TXA_  4
#define NT_   2048
#define L_    16
#define T_    2063                       // NT + L - 1
#define MT_   16                         // m-tile per workgroup
#define NMT_  ((T_ + MT_ - 1) / MT_)     // 129 tiles
#define QM_   4                          // m-rows per TDM quarter-descriptor
#define NQ_   (MT_ / QM_)                // 4 descriptors per tile

// ---------------- LDS layout ----------------
// Four quarter tiles, each [c=16][a=4][m=4][l=16] f32 (l fastest) filled by one
// TDM descriptor with pad_enable (2 dwords per 256 dwords). Per-quarter
// footprint: 4096 data + 32 pad dwords = 16512 B. The pad skews the c-stride
// to 258 dwords so per-lane WMMA A-fetches stay 8B-aligned, <=2-way conflicted.
#define Q_DW      (RXA_ * TXA_ * QM_ * L_)     // 4096 dwords
#define Q_DW_PAD  (Q_DW + 2 * (Q_DW / 256))    // 4128 dwords
#define Q_B       (Q_DW_PAD * 4)               // 16512 bytes
#define XW_OFF_B  (NQ_ * Q_B + 256)            // x-window offset (66304)
#define SMEM_B    (XW_OFF_B + TXA_ * 32 * 4)   // ~66.8 KB dynamic LDS

typedef float     v2f __attribute__((ext_vector_type(2)));
typedef float     v8f __attribute__((ext_vector_type(8)));
typedef unsigned  v4u __attribute__((ext_vector_type(4)));
typedef int       v4i __attribute__((ext_vector_type(4)));
typedef int       v8i __attribute__((ext_vector_type(8)));

// These evaluate per compilation pass: true on the gfx1250 device pass
// (proven in round 3), false on the x86 host pass (which must still parse
// this TU but never executes the kernel).
#define HAS_TDM_ \
  (__has_builtin(__builtin_amdgcn_tensor_load_to_lds) && \
   __has_builtin(__builtin_amdgcn_s_wait_tensorcnt))
#define HAS_WMMA_ __has_builtin(__builtin_amdgcn_wmma_f32_16x16x4_f32)

// Hard guarantee for the DEVICE pass only: if the gfx1250 pass ever loses
// these builtins, fail loudly instead of silently taking a fallback.
#if defined(__HIP_DEVICE_COMPILE__)
#if !HAS_TDM_
#error "DEVICE PASS: tensor_load_to_lds / s_wait_tensorcnt builtins ABSENT"
#endif
#if !HAS_WMMA_
#error "DEVICE PASS: wmma_f32_16x16x4_f32 builtin ABSENT"
#endif
#endif

// Per-quarter compute. QQ is a template parameter so the s_wait_tensorcnt
// immediate is an integral constant expression at the frontend.
template <int QQ>
__device__ __forceinline__ void compute_quarter(const float* __restrict__ hs,
                                                const float* __restrict__ xw,
                                                int cA, int hh,
                                                v8f& acc0, v8f& acc1) {
#if HAS_TDM_
  // Tensor ops from one wave complete in-order: quarter QQ is resident once
  // TENSORcnt <= (NQ_-1-QQ). Compute overlaps the remaining DMAs.
  __builtin_amdgcn_s_wait_tensorcnt(NQ_ - 1 - QQ);
#endif
#if HAS_WMMA_
#pragma unroll 2                          // pair even/odd j -> dual chains + 2addr fetches
  for (int jm = 0; jm < QM_; ++jm) {
    const int j = QQ * QM_ + jm;
#pragma unroll
    for (int q = 0; q < 16; ++q) {
      const int k0 = 4 * q + 2 * hh;      // this lane's first K of the chunk
      const int a  = k0 >> 4;
      const int l  = k0 & 15;
      // A[c, K] = h[c, m0+j, k0..k0+1]  (documented 16x4 f32 A layout)
      const int p = ((cA * TXA_ + a) * QM_ + jm) * L_ + l;
      const v2f av =
          *reinterpret_cast<const v2f*>(hs + QQ * Q_DW_PAD + p + 2 * (p >> 8));
      // B[K, n] one-hot in column j: value x[b, a, (m0+j) - l]
      const float x0 = xw[a * 32 + (j + (L_ - 1) - l)];
      const float x1 = xw[a * 32 + (j + (L_ - 1) - l - 1)];
      const bool sel = (cA == j);
      v2f bv;
      bv.x = sel ? x0 : 0.0f;
      bv.y = sel ? x1 : 0.0f;
      if (jm & 1)
        acc1 = __builtin_amdgcn_wmma_f32_16x16x4_f32(
            false, av, false, bv, (short)0, acc1, false, false);
      else
        acc0 = __builtin_amdgcn_wmma_f32_16x16x4_f32(
            false, av, false, bv, (short)0, acc0, false, false);
    }
  }
#else
  (void)hs; (void)xw; (void)cA; (void)hh; (void)acc0; (void)acc1;
#endif
}

__global__ __launch_bounds__(32)
void atc_kernel(const float* __restrict__ x, const float* __restrict__ h,
                float* __restrict__ y) {
  extern __shared__ char smem[];
  float* hs = reinterpret_cast<float*>(smem);
  float* xw = reinterpret_cast<float*>(smem + XW_OFF_B);

  const int lane = threadIdx.x;            // wave32: one wave per workgroup
  const int b    = blockIdx.y;
  const int m0   = blockIdx.x * MT_;

  // h_time[b, r=0, c, t=0, a, m, l]; strides: l=1, m=16, a=T*16, c=4*T*16
  const float* hb = h + (size_t)b * ((size_t)RXA_ * TXA_ * T_ * L_);

  // ---- stage x window: xw[a][i] = x[b,a, m0-15+i] for i=0..30, 0 outside [0,NT) ----
  for (int idx = lane; idx < TXA_ * 32; idx += 32) {
    const int a = idx >> 5, i = idx & 31;
    const int t = m0 - (L_ - 1) + i;
    float v = 0.0f;
    if (i < 2 * L_ - 1 && t >= 0 && t < NT_)
      v = x[((size_t)b * TXA_ + a) * NT_ + t];
    xw[idx] = v;
  }

#if HAS_TDM_
  // ---- issue 4 TDM quarter-descriptors back-to-back (in-order completion) ----
  const unsigned lds_base =
      (unsigned)(size_t)(__attribute__((address_space(3))) char*)smem;
  for (int qq = 0; qq < NQ_; ++qq) {
    const int mq = m0 + qq * QM_;                        // first m of quarter
    const uint64_t ga = (uint64_t)(uintptr_t)(hb + (size_t)mq * L_);
    const int dim1 = T_ - mq;            // remaining m rows; HW zero-fills OOB

    union { v4u v; unsigned u[4]; } g0 = {};
    g0.u[0] = 1u;                                            // count=1, normal mode
    g0.u[1] = lds_base + (unsigned)(qq * Q_B);               // lds_addr (bytes)
    g0.u[2] = (unsigned)ga;                                  // global_addr lo32
    g0.u[3] = (unsigned)((ga >> 32) & 0x01FFFFFFu) | (2u << 30); // hi25 | type=2

    union { v8i v; int i[8]; } g1 = {};
    // data_size=4B(2), pad_enable, pad_interval=256dw(7), pad_amount=2dw(1)
    g1.i[0] = (2 << 16) | (1 << 20) | (7 << 22) | (1 << 25);
    g1.i[1] = (L_ << 16);                 // tensor_dim0 = 16 (l)
    g1.i[2] = (dim1 & 0xFFFF) << 16;      // tensor_dim1 = T - mq (m), lo16
    g1.i[3] = (L_ << 16);                 // dim1 hi16=0 | tile_dim0 = 16
    g1.i[4] = QM_ | (TXA_ << 16);         // tile_dim1 = 4 (m) | tile_dim2 = 4 (a)
    g1.i[5] = L_;                         // tensor_dim0_stride = 16
    g1.i[6] = (L_ << 16);                 // stride0 hi=0 | tensor_dim1_stride = 16
    g1.i[7] = 0;

    union { v4i v; int i[4]; } g2 = {};
    g2.i[0] = TXA_;                       // tensor_dim2 = 4 (a)
    g2.i[1] = RXA_;                       // tensor_dim3 = 16 (c)
    g2.i[2] = T_ * L_;                    // tensor_dim2_stride = 33008
    g2.i[3] = (RXA_ << 16);               // stride2 hi=0 | tile_dim3 = 16 (c)

    union { v4i v; int i[4]; } g3 = {};
    g3.i[0] = TXA_ * T_ * L_;             // tensor_dim3_stride = 132032
    // tensor_dim4 / tile_dim4 = 0 (unused)

#if __clang_major__ >= 23
    union { v8i v; int i[8]; } g4 = {};   // 6-arg variant (amdgpu-toolchain)
    __builtin_amdgcn_tensor_load_to_lds(g0.v, g1.v, g2.v, g3.v, g4.v, 0);
#else
    __builtin_amdgcn_tensor_load_to_lds(g0.v, g1.v, g2.v, g3.v, 0);
#endif
  }
#else
  // Fallback: coalesced manual staging into the same quarter-tile LDS layout.
  for (int pos = lane; pos < NQ_ * Q_DW; pos += 32) {
    const int qq  = pos >> 12;            // quarter
    const int rem = pos & (Q_DW - 1);     // c*256 + a*64 + jm*16 + l
    const int ca  = rem >> 6;             // c*4 + a
    const int jm  = (rem >> 4) & 3;
    const int mm  = m0 + qq * QM_ + jm;
    float v = 0.0f;
    if (mm < T_)
      v = hb[(size_t)ca * (T_ * L_) + (size_t)mm * L_ + (rem & 15)];
    hs[qq * Q_DW_PAD + rem + 2 * (rem >> 8)] = v;
  }
#endif

  // ---- compute: D[c, j] = sum_k H[c, m0+j, k] * X[m0+j, k], k = a*16 + l ----
  // Even/odd-j accumulators halve the WMMA D->C RAW chain; one-hot column
  // contributions are disjoint, so final D = acc0 + acc1.
  union { v8f v; float f[8]; } acc0, acc1;
#pragma unroll
  for (int r = 0; r < 8; ++r) { acc0.f[r] = 0.0f; acc1.f[r] = 0.0f; }

  const int cA = lane & 15;               // A-matrix row (c) / B column id
  const int hh = lane >> 4;               // lane half: K rows {0,1} vs {2,3}

#if HAS_WMMA_
  compute_quarter<0>(hs, xw, cA, hh, acc0.v, acc1.v);   // waits tensorcnt<=3
  compute_quarter<1>(hs, xw, cA, hh, acc0.v, acc1.v);   // waits tensorcnt<=2
  compute_quarter<2>(hs, xw, cA, hh, acc0.v, acc1.v);   // waits tensorcnt<=1
  compute_quarter<3>(hs, xw, cA, hh, acc0.v, acc1.v);   // waits tensorcnt<=0
#else
  // VALU fallback (also what the host pass parses): lane (cA,hh) -> col cA
  {
    const int qq = cA >> 2, jm = cA & 3;
    for (int r = 0; r < 8; ++r) {
      const int c = r + 8 * hh;
      float s = 0.0f;
      for (int k = 0; k < 64; ++k) {
        const int a = k >> 4, l = k & 15;
        const int p = ((c * TXA_ + a) * QM_ + jm) * L_ + l;
        s += hs[qq * Q_DW_PAD + p + 2 * (p >> 8)] *
             xw[a * 32 + (cA + (L_ - 1) - l)];
      }
      acc0.f[r] = s;
    }
  }
#endif

  // ---- store: D layout VGPR r, lanes0-15: (c=r, j=lane); lanes16-31: (c=r+8, j=lane-16)
  const int m = m0 + cA;
  if (m < T_) {
#pragma unroll
    for (int r = 0; r < 8; ++r)
      y[((size_t)b * RXA_ + (r + 8 * hh)) * T_ + m] = acc0.f[r] + acc1.f[r];
  }
}

extern "C" void kernel_launch(void* const* d_in, const int* in_sizes, int n_in,
                              void* d_out, int out_size, void* d_ws, size_t ws_size,
                              hipStream_t stream) {
  (void)in_sizes; (void)n_in; (void)out_size; (void)d_ws; (void)ws_size;
  const float* x = (const float*)d_in[0];
  const float* h = (const float*)d_in[1];
  // d_in[2] (Toeplitz index table g) is not needed: indices are analytic.
  float* y = (float*)d_out;
  dim3 grid(NMT_, B_, 1);
  dim3 block(32, 1, 1);
  atc_kernel<<<grid, block, SMEM_B, stream>>>(x, h, y);
}